// GCN_70944269795814
// MI455X (gfx1250) — compile-verified
//
#include <hip/hip_runtime.h>
#include <hip/hip_bf16.h>

typedef __attribute__((ext_vector_type(2))) float v2f;
typedef __attribute__((ext_vector_type(8))) float v8f;

#define N_NODES 100000
#define N_EDGES 1600000
#define FEAT 128
#define EMB 32
#define N_GRAPHS 512

// ---------------------------------------------------------------------------
// fill kernel: p[i] = v for i < n
// ---------------------------------------------------------------------------
__global__ void gcn_fill_kernel(float* __restrict__ p, float v, int n) {
    int i = blockIdx.x * blockDim.x + threadIdx.x;
    if (i < n) p[i] = v;
}

// ---------------------------------------------------------------------------
// degree accumulation: deg[dst[e]] += 1  (deg pre-filled with 1.0 for self loop)
// ---------------------------------------------------------------------------
__global__ void gcn_deg_kernel(const int* __restrict__ dst, float* __restrict__ deg) {
    int e = blockIdx.x * blockDim.x + threadIdx.x;
    if (e < N_EDGES) atomicAdd(&deg[dst[e]], 1.0f);
}

// deg -> d^{-1/2}, in place
__global__ void gcn_rsqrt_kernel(float* __restrict__ deg) {
    int i = blockIdx.x * blockDim.x + threadIdx.x;
    if (i < N_NODES) deg[i] = rsqrtf(deg[i]);
}

// ---------------------------------------------------------------------------
// H = X @ W  via V_WMMA_F32_16X16X4_F32.  X: [n,K] row-major, W: [K,EMB] row-
// major, H: [n,EMB].  One wave per 16 rows; two 16-wide N tiles (EMB == 32).
// A layout: lane&15 = M, VGPRs hold K = koff, koff+1, koff = 2*(lane>>4).
// B layout: lane&15 = N, VGPRs hold K = koff, koff+1 (same koff).
// C/D layout: VGPR r -> row (r + 8*(lane>>4)), col = lane&15.
// ---------------------------------------------------------------------------
template <int K>
__global__ void gcn_wmma_gemm_kernel(const float* __restrict__ X,
                                     const float* __restrict__ W,
                                     float* __restrict__ H,
                                     int n_rows) {
    const int lane = threadIdx.x & 31;
    const int wave = (blockIdx.x * blockDim.x + threadIdx.x) >> 5;
    const int rowbase = wave * 16;
    if (rowbase >= n_rows) return;              // wave-uniform: EXEC stays all-1s

    const int m    = lane & 15;                 // M for A, N for B, col for C
    const int koff = (lane >> 4) << 1;          // 0 or 2

    const float* xrow = X + (size_t)(rowbase + m) * K;

    v8f c0 = {};
    v8f c1 = {};
    for (int k0 = 0; k0 < K; k0 += 4) {
        v2f a;
        a.x = xrow[k0 + koff];
        a.y = xrow[k0 + koff + 1];

        const float* wr0 = W + (size_t)(k0 + koff) * EMB;
        const float* wr1 = wr0 + EMB;
        v2f b0, b1;
        b0.x = wr0[m];      b0.y = wr1[m];       // N tile 0 (cols 0..15)
        b1.x = wr0[m + 16]; b1.y = wr1[m + 16];  // N tile 1 (cols 16..31)

        c0 = __builtin_amdgcn_wmma_f32_16x16x4_f32(
                 false, a, false, b0, (short)0, c0, false, false);
        c1 = __builtin_amdgcn_wmma_f32_16x16x4_f32(
                 false, a, false, b1, (short)0, c1, false, false);
    }

    const int rhi = (lane >> 4) << 3;           // 0 or 8
#pragma unroll
    for (int r = 0; r < 8; ++r) {
        const size_t row = (size_t)(rowbase + rhi + r);
        H[row * EMB + m]      = c0[r];
        H[row * EMB + 16 + m] = c1[r];
    }
}

// ---------------------------------------------------------------------------
// edge scatter: AGG[dst] += H[src] * (dis[src]*dis[dst]); 8 threads per edge,
// float4 gather + 4 global f32 atomics each.
// ---------------------------------------------------------------------------
__global__ void gcn_scatter_kernel(const float* __restrict__ H,
                                   const int* __restrict__ src,
                                   const int* __restrict__ dst,
                                   const float* __restrict__ dis,
                                   float* __restrict__ AGG) {
    long long tid = (long long)blockIdx.x * blockDim.x + threadIdx.x;
    long long e = tid >> 3;
    if (e >= N_EDGES) return;
    const int c = (int)(tid & 7) * 4;
    const int s = src[e];
    const int d = dst[e];
    const float norm = dis[s] * dis[d];
    const float4 hv = *(const float4*)(H + (size_t)s * EMB + c);
    float* out = AGG + (size_t)d * EMB + c;
    atomicAdd(out + 0, hv.x * norm);
    atomicAdd(out + 1, hv.y * norm);
    atomicAdd(out + 2, hv.z * norm);
    atomicAdd(out + 3, hv.w * norm);
}

// ---------------------------------------------------------------------------
// OUT = relu(AGG + H * dis^2 + bias)   (elementwise over N_NODES x EMB)
// ---------------------------------------------------------------------------
__global__ void gcn_combine_kernel(const float* __restrict__ AGG,
                                   const float* __restrict__ H,
                                   const float* __restrict__ dis,
                                   const float* __restrict__ bias,
                                   float* __restrict__ OUT) {
    int i = blockIdx.x * blockDim.x + threadIdx.x;
    if (i >= N_NODES * EMB) return;
    const int node = i >> 5;
    const int j = i & 31;
    const float dii = dis[node];
    const float v = AGG[i] + H[i] * (dii * dii) + bias[j];
    OUT[i] = fmaxf(v, 0.0f);
}

// ---------------------------------------------------------------------------
// pooled segment sum + per-graph counts
// ---------------------------------------------------------------------------
__global__ void gcn_pool_kernel(const float* __restrict__ X,
                                const int* __restrict__ batch,
                                float* __restrict__ SUM,
                                float* __restrict__ CNT) {
    int i = blockIdx.x * blockDim.x + threadIdx.x;
    if (i >= N_NODES * EMB) return;
    const int node = i >> 5;
    const int j = i & 31;
    const int g = batch[node];
    atomicAdd(&SUM[(size_t)g * EMB + j], X[i]);
    if (j == 0) atomicAdd(&CNT[g], 1.0f);
}

// ---------------------------------------------------------------------------
// out[g] = (SUM[g]/max(CNT[g],1)) . Wo + bo
// ---------------------------------------------------------------------------
__global__ void gcn_final_kernel(const float* __restrict__ SUM,
                                 const float* __restrict__ CNT,
                                 const float* __restrict__ Wo,
                                 const float* __restrict__ bo,
                                 float* __restrict__ out) {
    int g = blockIdx.x * blockDim.x + threadIdx.x;
    if (g >= N_GRAPHS) return;
    const float inv = 1.0f / fmaxf(CNT[g], 1.0f);
    float acc = 0.0f;
#pragma unroll
    for (int j = 0; j < EMB; ++j) acc += SUM[(size_t)g * EMB + j] * Wo[j];
    out[g] = acc * inv + bo[0];
}

static inline size_t align256(size_t x) { return (x + 255) & ~(size_t)255; }

extern "C" void kernel_launch(void* const* d_in, const int* in_sizes, int n_in,
                              void* d_out, int out_size, void* d_ws, size_t ws_size,
                              hipStream_t stream) {
    // inputs (setup_inputs order): x, edge_index, batch_index, W1, b1, W2, b2, Wo, bo
    const float* x     = (const float*)d_in[0];
    const int*   eidx  = (const int*)d_in[1];
    const int*   batch = (const int*)d_in[2];
    const float* W1    = (const float*)d_in[3];
    const float* b1    = (const float*)d_in[4];
    const float* W2    = (const float*)d_in[5];
    const float* b2    = (const float*)d_in[6];
    const float* Wo    = (const float*)d_in[7];
    const float* bo    = (const float*)d_in[8];
    float* out = (float*)d_out;

    const int* src = eidx;             // edge_index[0]
    const int* dst = eidx + N_EDGES;   // edge_index[1]

    // workspace carve-up
    char* ws = (char*)d_ws;
    size_t off = 0;
    float* dis  = (float*)(ws + off); off = align256(off + (size_t)N_NODES * 4);
    float* bufA = (float*)(ws + off); off = align256(off + (size_t)N_NODES * EMB * 4);
    float* bufB = (float*)(ws + off); off = align256(off + (size_t)N_NODES * EMB * 4);
    float* SUM  = (float*)(ws + off); off = align256(off + (size_t)N_GRAPHS * EMB * 4);
    float* CNT  = (float*)(ws + off); off = align256(off + (size_t)N_GRAPHS * 4);
    (void)ws_size;

    const int NE = N_NODES * EMB;                     // 3.2M elems
    const int TB = 256;

    // 1. degree -> d^{-1/2}
    gcn_fill_kernel<<<(N_NODES + TB - 1) / TB, TB, 0, stream>>>(dis, 1.0f, N_NODES);
    gcn_deg_kernel<<<(N_EDGES + TB - 1) / TB, TB, 0, stream>>>(dst, dis);
    gcn_rsqrt_kernel<<<(N_NODES + TB - 1) / TB, TB, 0, stream>>>(dis);

    // GEMM launch geometry: one wave per 16 rows, 4 waves (128 threads) / block
    const int n_waves  = N_NODES / 16;                // 6250 exact
    const int gemm_blk = (n_waves + 3) / 4;

    // 2. layer 1: h1 = x @ W1  -> bufA
    gcn_wmma_gemm_kernel<FEAT><<<gemm_blk, 128, 0, stream>>>(x, W1, bufA, N_NODES);
    // agg1 = 0 -> bufB
    gcn_fill_kernel<<<(NE + TB - 1) / TB, TB, 0, stream>>>(bufB, 0.0f, NE);
    // scatter edges
    {
        long long nthr = (long long)N_EDGES * 8;
        gcn_scatter_kernel<<<(int)((nthr + TB - 1) / TB), TB, 0, stream>>>(bufA, src, dst, dis, bufB);
    }
    // out1 = relu(agg1 + h1*dis^2 + b1) -> bufA (elementwise, safe in place)
    gcn_combine_kernel<<<(NE + TB - 1) / TB, TB, 0, stream>>>(bufB, bufA, dis, b1, bufA);

    // 3. layer 2: h2 = out1 @ W2 -> bufB
    gcn_wmma_gemm_kernel<EMB><<<gemm_blk, 128, 0, stream>>>(bufA, W2, bufB, N_NODES);
    // agg2 = 0 -> bufA (out1 no longer needed)
    gcn_fill_kernel<<<(NE + TB - 1) / TB, TB, 0, stream>>>(bufA, 0.0f, NE);
    {
        long long nthr = (long long)N_EDGES * 8;
        gcn_scatter_kernel<<<(int)((nthr + TB - 1) / TB), TB, 0, stream>>>(bufB, src, dst, dis, bufA);
    }
    // out2 = relu(agg2 + h2*dis^2 + b2) -> bufB
    gcn_combine_kernel<<<(NE + TB - 1) / TB, TB, 0, stream>>>(bufA, bufB, dis, b2, bufB);

    // 4. global mean pool + readout
    gcn_fill_kernel<<<(N_GRAPHS * EMB + TB - 1) / TB, TB, 0, stream>>>(SUM, 0.0f, N_GRAPHS * EMB);
    gcn_fill_kernel<<<(N_GRAPHS + TB - 1) / TB, TB, 0, stream>>>(CNT, 0.0f, N_GRAPHS);
    gcn_pool_kernel<<<(NE + TB - 1) / TB, TB, 0, stream>>>(bufB, batch, SUM, CNT);
    gcn_final_kernel<<<(N_GRAPHS + TB - 1) / TB, TB, 0, stream>>>(SUM, CNT, Wo, bo, out);
}